// PointNetSaModule_2697239462400
// MI455X (gfx1250) — compile-verified
//
#include <hip/hip_runtime.h>

// ---- problem constants (match reference) ----
#define BSZ   2
#define HDIM  64
#define WDIM  1024
#define CDIM  64
#define HW    (HDIM*WDIM)     // 65536
#define NPTS  16384
#define KNBR  32
#define TN    2               // sample points per block
#define MROWS (TN*KNBR)       // 64 rows = 4 M-tiles
#define C0    67              // input channels (3 + 64)
#define PITCH0 68             // padded pitch (== 4 mod 64 -> conflict-free)
#define CO3   128
#define PITCH3 132

// LDS arena layout (floats):
//  [0      .. 4352)  feat (stage1 in), later act2 (stage2 out / stage3 in)
//  [4352   .. 8704)  act1 (stage1 out / stage2 in)
//  [4352   ..12800)  act3 (stage3 out) -- overlays act1, which is dead then
#define OFF_FEAT 0
#define OFF_ACT1 4352
#define OFF_ACT3 4352
#define ARENA    (4352 + MROWS*PITCH3)   // 12800 floats = 51200 B

typedef __attribute__((ext_vector_type(2))) float v2f;
typedef __attribute__((ext_vector_type(8))) float v8f;

// One GEMM stage: out = relu(in @ (W*g) + (b*g+be)), 16x16 tiles via
// V_WMMA_F32_16X16X4_F32. B-fragments (weights) held in VGPRs, A from LDS.
template<int KSTEPS>
__device__ __forceinline__ void gemm_stage(
    const float* __restrict__ Wg, const float* __restrict__ bb,
    const float* __restrict__ gg, const float* __restrict__ bee,
    int kin, int co,
    const float* __restrict__ in_lds, int in_pitch,
    float* __restrict__ out_lds, int out_pitch,
    int ntile, int m_lo, int m_hi, int lane)
{
    const int   nc   = (lane & 15) + ntile * 16;   // output column
    const int   kb   = (lane >> 4) * 2;            // k sub-offset for this half-wave
    const float gn   = gg[nc];
    const float bias = bb[nc] * gn + bee[nc];

    // B fragments: element (k = 4s + kb + v, n = nc), folded with BN scale
    v2f bfrag[KSTEPS];
#pragma unroll
    for (int s = 0; s < KSTEPS; ++s) {
        const int k0 = 4 * s + kb;
        bfrag[s].x = (k0     < kin) ? Wg[(k0    ) * co + nc] * gn : 0.0f;
        bfrag[s].y = (k0 + 1 < kin) ? Wg[(k0 + 1) * co + nc] * gn : 0.0f;
    }

    for (int mt = m_lo; mt < m_hi; ++mt) {
        const float* arow = in_lds + (mt * 16 + (lane & 15)) * in_pitch + kb;
        v8f acc = {};
#pragma unroll
        for (int s = 0; s < KSTEPS; ++s) {
            v2f a = *(const v2f*)(arow + 4 * s);   // 8B-aligned LDS load
            acc = __builtin_amdgcn_wmma_f32_16x16x4_f32(
                false, a, false, bfrag[s], (short)0, acc, false, false);
        }
        // D layout: m = r + 8*(lane>>4), n = lane&15
        float* orow = out_lds + (mt * 16 + ((lane >> 4) << 3)) * out_pitch + nc;
#pragma unroll
        for (int r = 0; r < 8; ++r) {
            float v = acc[r] + bias;
            orow[r * out_pitch] = fmaxf(v, 0.0f);
        }
    }
}

__launch_bounds__(256, 1)
__global__ void pointnet_sa_kernel(
    const float* __restrict__ xyz,    // [B, HW, 3]
    const float* __restrict__ pts,    // [B, HW, C]
    const float* __restrict__ xyzs,   // [B, N, 3]
    const int*   __restrict__ nidx,   // [B, N, K]
    const float* __restrict__ vmask,  // [B, N, K, 1]
    const float* __restrict__ W0, const float* __restrict__ b0,
    const float* __restrict__ g0, const float* __restrict__ be0,
    const float* __restrict__ W1, const float* __restrict__ b1,
    const float* __restrict__ g1, const float* __restrict__ be1,
    const float* __restrict__ W2, const float* __restrict__ b2,
    const float* __restrict__ g2, const float* __restrict__ be2,
    float* __restrict__ out)          // [2 * B*N*128] (tuple: flat + reshape)
{
    __shared__ float arena[ARENA];
    __shared__ int   s_idx[MROWS];
    __shared__ float s_msk[MROWS];
    __shared__ float s_nxyz[TN * 3];

    const int tid  = threadIdx.x;
    const int lane = tid & 31;
    const int wv   = tid >> 5;            // 8 waves
    const int blk  = blockIdx.x;
    const int b    = blk / (NPTS / TN);
    const int n0   = (blk % (NPTS / TN)) * TN;

    // ---- row metadata: neighbor index, mask, query xyz ----
    if (tid < MROWS) {
        const int flat = (b * NPTS + n0) * KNBR + tid;  // rows are contiguous
        s_idx[tid] = nidx[flat];
        s_msk[tid] = vmask[flat];
    }
    if (tid < TN * 3) s_nxyz[tid] = xyzs[(b * NPTS + n0) * 3 + tid];
    __syncthreads();

    // ---- gather + concat: feat[r][c] = (c<3 ? xyz*m - nxyz : pts*m), pad c=67 ----
    {
        const float* xb = xyz + (size_t)b * HW * 3;
        const float* pb = pts + (size_t)b * HW * CDIM;
        for (int e = tid; e < MROWS * PITCH0; e += 256) {
            const int r = e / PITCH0, c = e - r * PITCH0;
            const int gi = s_idx[r];
            const float m = s_msk[r];
            float v;
            if (c < 3)       v = xb[gi * 3 + c] * m - s_nxyz[(r >> 5) * 3 + c];
            else if (c < C0) v = pb[gi * CDIM + (c - 3)] * m;
            else             v = 0.0f;
            arena[OFF_FEAT + r * PITCH0 + c] = v;
        }
    }
    __syncthreads();

    // ---- stage 1: [64x67] @ W0[67x64]  (17 k-steps, padded) ----
    {
        const int j    = wv & 3;                 // 4 N-tiles
        const int m_lo = (wv < 4) ? 0 : 2;       // split 4 M-tiles across wave pairs
        gemm_stage<17>(W0, b0, g0, be0, C0, 64,
                       arena + OFF_FEAT, PITCH0,
                       arena + OFF_ACT1, PITCH0,
                       j, m_lo, m_lo + 2, lane);
    }
    __syncthreads();

    // ---- stage 2: [64x64] @ W1[64x64] -> act2 (reuses feat slot) ----
    {
        const int j    = wv & 3;
        const int m_lo = (wv < 4) ? 0 : 2;
        gemm_stage<16>(W1, b1, g1, be1, 64, 64,
                       arena + OFF_ACT1, PITCH0,
                       arena + OFF_FEAT, PITCH0,
                       j, m_lo, m_lo + 2, lane);
    }
    __syncthreads();

    // ---- stage 3: [64x64] @ W2[64x128] -> act3 (overlays act1) ----
    gemm_stage<16>(W2, b2, g2, be2, 64, CO3,
                   arena + OFF_FEAT, PITCH0,
                   arena + OFF_ACT3, PITCH3,
                   wv /* 8 N-tiles */, 0, 4, lane);
    __syncthreads();

    // ---- max-pool over K=32 neighbors, dual store (tuple outputs) ----
    for (int e = tid; e < TN * CO3; e += 256) {
        const int nl = e / CO3, c = e - nl * CO3;
        const float* col = arena + OFF_ACT3 + (nl * KNBR) * PITCH3 + c;
        float mv = col[0];
#pragma unroll
        for (int r = 1; r < KNBR; ++r) mv = fmaxf(mv, col[r * PITCH3]);
        const size_t o = ((size_t)b * NPTS + n0 + nl) * CO3 + c;
        out[o] = mv;
        out[(size_t)BSZ * NPTS * CO3 + o] = mv;   // reshaped copy = same layout
    }
}

extern "C" void kernel_launch(void* const* d_in, const int* in_sizes, int n_in,
                              void* d_out, int out_size, void* d_ws, size_t ws_size,
                              hipStream_t stream) {
    (void)in_sizes; (void)n_in; (void)out_size; (void)d_ws; (void)ws_size;
    const float* xyz  = (const float*)d_in[0];
    const float* pts  = (const float*)d_in[1];
    const float* xyzs = (const float*)d_in[2];
    const int*   nidx = (const int*)  d_in[3];
    const float* vmsk = (const float*)d_in[4];
    const float* W0 = (const float*)d_in[5],  *b0 = (const float*)d_in[6];
    const float* g0 = (const float*)d_in[7],  *be0 = (const float*)d_in[8];
    const float* W1 = (const float*)d_in[9],  *b1 = (const float*)d_in[10];
    const float* g1 = (const float*)d_in[11], *be1 = (const float*)d_in[12];
    const float* W2 = (const float*)d_in[13], *b2 = (const float*)d_in[14];
    const float* g2 = (const float*)d_in[15], *be2 = (const float*)d_in[16];
    float* out = (float*)d_out;

    dim3 grid(BSZ * (NPTS / TN));   // 16384 blocks
    pointnet_sa_kernel<<<grid, 256, 0, stream>>>(
        xyz, pts, xyzs, nidx, vmsk,
        W0, b0, g0, be0, W1, b1, g1, be1, W2, b2, g2, be2, out);
}